// SpanRepLayer_82480551952798
// MI455X (gfx1250) — compile-verified
//
#include <hip/hip_runtime.h>
#include <stdint.h>

// ---------------------------------------------------------------------------
// SpanRepLayer for MI455X (gfx1250).
// Gather start/end rows + masked max-pool over inner window per span.
// Memory-bound: uses CDNA5 async global->LDS copies (ASYNCcnt) to stage each
// span's contiguous <=16-row window into LDS, then computes from LDS.
// ---------------------------------------------------------------------------

#ifndef __has_builtin
#define __has_builtin(x) 0
#endif

typedef int v4i __attribute__((ext_vector_type(4)));
typedef __attribute__((address_space(1))) v4i gas_v4i;   // global int4
typedef __attribute__((address_space(3))) v4i las_v4i;   // LDS int4
typedef __attribute__((address_space(3))) char las_char; // LDS byte

__device__ __forceinline__ void async_copy_b128(const void* gsrc, void* lds_dst) {
#if __has_builtin(__builtin_amdgcn_global_load_async_to_lds_b128)
  __builtin_amdgcn_global_load_async_to_lds_b128(
      (gas_v4i*)gsrc, (las_v4i*)lds_dst, /*offset=*/0, /*cpol=*/0);
#else
  // vdst = LDS byte address, vaddr = 64-bit global address, saddr = off
  uint32_t loff = (uint32_t)(uintptr_t)(las_char*)lds_dst;
  uint64_t ga   = (uint64_t)(uintptr_t)gsrc;
  asm volatile("global_load_async_to_lds_b128 %0, %1, off"
               :: "v"(loff), "v"(ga) : "memory");
#endif
}

__device__ __forceinline__ void wait_async_zero() {
#if __has_builtin(__builtin_amdgcn_s_wait_asynccnt)
  __builtin_amdgcn_s_wait_asynccnt(0);
#else
  asm volatile("s_wait_asynccnt 0" ::: "memory");
#endif
  asm volatile("" ::: "memory");  // keep LDS reads ordered after the wait
}

// Problem dimensions (fixed by the reference).
static constexpr int Bc    = 2;
static constexpr int NSc   = 512;
static constexpr int Sc    = 1024;
static constexpr int Hc    = 128;
static constexpr int MAXWc = 16;

static constexpr int ROWB       = Hc * 4;          // 512 B per token row
static constexpr int WAVE_BYTES = MAXWc * ROWB;    // 8 KB staging per wave
static constexpr int WPB        = 8;               // waves per block (256 thr)

__global__ __launch_bounds__(32 * WPB)
void span_rep_kernel(const float* __restrict__ token_reps,
                     const int*   __restrict__ span_ids,
                     const unsigned char* __restrict__ span_masks,
                     float* __restrict__ out)
{
  __shared__ __align__(16) char smem[WPB * WAVE_BYTES];   // 64 KB / block

  const int lane = threadIdx.x & 31;
  const int wave = threadIdx.x >> 5;
  const int g    = blockIdx.x * WPB + wave;               // flat span id
  if (g >= Bc * NSc) return;

  const int  b     = g / NSc;
  const int  start = span_ids[2 * g];
  const int  end   = span_ids[2 * g + 1] - 1;             // inclusive end
  const bool valid = span_masks[g] != 0;                  // jax bool = 1 byte

  float* outp = out + (size_t)g * (3 * Hc);

  if (!valid) {
    // Reference zeroes invalid spans entirely.
    const float4 z = make_float4(0.f, 0.f, 0.f, 0.f);
    *(float4*)(outp +          4 * lane) = z;
    *(float4*)(outp + Hc     + 4 * lane) = z;
    *(float4*)(outp + 2 * Hc + 4 * lane) = z;
    return;
  }

  const int len = end - start + 1;                        // 1..16
  const float* win = token_reps + ((size_t)b * Sc + start) * Hc;
  char* myLds = smem + wave * WAVE_BYTES;

  // Stage the contiguous window [start..end] (len * 512 B) into LDS.
  // Each lane moves 16 B per row -> one fully coalesced 512 B async copy/row.
  for (int r = 0; r < len; ++r) {
    async_copy_b128(win + r * Hc + 4 * lane, myLds + r * ROWB + 16 * lane);
  }
  wait_async_zero();

  const float4* rows = (const float4*)myLds;              // 32 float4 per row
  const float4  sv   = rows[lane];                        // start row
  const float4  ev   = rows[(len - 1) * 32 + lane];       // end row

  // Masked max-pool over inner rows [1, len-2]; NEG_LIMIT seed matches the
  // reference's jnp.where(..., NEG_LIMIT) fill exactly.
  float4 acc = make_float4(-1e9f, -1e9f, -1e9f, -1e9f);
  for (int r = 1; r < len - 1; ++r) {
    const float4 v = rows[r * 32 + lane];
    acc.x = fmaxf(acc.x, v.x);
    acc.y = fmaxf(acc.y, v.y);
    acc.z = fmaxf(acc.z, v.z);
    acc.w = fmaxf(acc.w, v.w);
  }
  const float4 inner = (len >= 3) ? acc : sv;             // has_inner fallback

  *(float4*)(outp +          4 * lane) = sv;
  *(float4*)(outp + Hc     + 4 * lane) = inner;
  *(float4*)(outp + 2 * Hc + 4 * lane) = ev;
}

extern "C" void kernel_launch(void* const* d_in, const int* in_sizes, int n_in,
                              void* d_out, int out_size, void* d_ws, size_t ws_size,
                              hipStream_t stream) {
  (void)in_sizes; (void)n_in; (void)out_size; (void)d_ws; (void)ws_size;
  const float*         token_reps = (const float*)d_in[0];
  const int*           span_ids   = (const int*)d_in[1];
  const unsigned char* span_masks = (const unsigned char*)d_in[2];
  float*               out        = (float*)d_out;

  const int total_spans = Bc * NSc;                       // 1024
  dim3 grid((total_spans + WPB - 1) / WPB);               // 128 blocks
  dim3 block(32 * WPB);                                   // 256 threads, 8 waves
  hipLaunchKernelGGL(span_rep_kernel, grid, block, 0, stream,
                     token_reps, span_ids, span_masks, out);
}